// MultiBoxLoss_83296595738888
// MI455X (gfx1250) — compile-verified
//
#include <hip/hip_runtime.h>
#include <hip/hip_bf16.h>
#include <math.h>

// MultiBox (SSD) loss for MI455X / gfx1250 (wave32).
// Roofline: dominated by one pass over predict_confs (64*8732*81*4 B = 181 MB
// -> ~7.8us at 23.3 TB/s). The CE kernel stages each 16-prior tile (5184 B,
// contiguous, 16B-aligned) into LDS via the Tensor Data Mover
// (tensor_load_to_lds + s_wait_tensorcnt; b128 fallback) and reduces the
// 81-class exp-sums with V_WMMA_F32_16X16X4_F32 (B = ones).
// Hard-negative top-K sum via exact bit-wise threshold search over an
// LDS-resident per-batch CE array. No atomics -> deterministic.

#define NB 64
#define NP 8732
#define NC 81
#define NO 16
#define VAR_XY 0.1f
#define VAR_WH 0.2f

typedef float v2f __attribute__((ext_vector_type(2)));
typedef float v8f __attribute__((ext_vector_type(8)));
typedef unsigned int u32x4 __attribute__((ext_vector_type(4)));
typedef int i32x8 __attribute__((ext_vector_type(8)));
typedef int i32x4 __attribute__((ext_vector_type(4)));

// ---------------------------------------------------------------------------
// Kernel 1: per-batch matching + positive labels + L1 loc-loss partial.
// ---------------------------------------------------------------------------
__global__ __launch_bounds__(256) void mbx_match(
    const float* __restrict__ plocs,    // [B,P,4]
    const float* __restrict__ tboxes,   // [B,O,4] xyxy
    const int*   __restrict__ tlabels,  // [B,O]
    const float* __restrict__ priors,   // [P,4] cxcywh
    int*   __restrict__ labels_ws,      // [B,P]
    int*   __restrict__ n_pos,          // [B]
    float* __restrict__ l1_part)        // [B]
{
  const int b = blockIdx.x, tid = threadIdx.x;
  __shared__ float s_box[NO][4];
  __shared__ float s_area[NO];
  __shared__ int   s_lab[NO];
  __shared__ int   s_objprior[NO];
  __shared__ float s_rf[NO * 256];
  __shared__ int   s_ri[NO * 256];

  if (tid < NO * 4) ((float*)s_box)[tid] = tboxes[b * NO * 4 + tid];
  if (tid < NO)     s_lab[tid] = tlabels[b * NO + tid];
  __syncthreads();
  if (tid < NO)
    s_area[tid] = (s_box[tid][2] - s_box[tid][0]) * (s_box[tid][3] - s_box[tid][1]);
  __syncthreads();

  const float4* pri4 = (const float4*)priors;
  const float4* pl4  = (const float4*)plocs;

  // ---- pass 1: per-object best prior (argmax over priors) ----
  float tb_iou[NO]; int tb_pri[NO];
#pragma unroll
  for (int o = 0; o < NO; ++o) { tb_iou[o] = -1.f; tb_pri[o] = 0; }

  for (int p = tid; p < NP; p += 256) {
    const float4 pr = pri4[p];
    const float px0 = pr.x - 0.5f*pr.z, py0 = pr.y - 0.5f*pr.w;
    const float px1 = pr.x + 0.5f*pr.z, py1 = pr.y + 0.5f*pr.w;
    const float pa = pr.z * pr.w;
#pragma unroll
    for (int o = 0; o < NO; ++o) {
      const float lx = fmaxf(s_box[o][0], px0), ly = fmaxf(s_box[o][1], py0);
      const float rx = fminf(s_box[o][2], px1), ry = fminf(s_box[o][3], py1);
      const float iw = fmaxf(rx - lx, 0.f), ih = fmaxf(ry - ly, 0.f);
      const float inter = iw * ih;
      const float iou = inter / (s_area[o] + pa - inter);
      if (iou > tb_iou[o]) { tb_iou[o] = iou; tb_pri[o] = p; }  // first max wins
    }
  }
#pragma unroll
  for (int o = 0; o < NO; ++o) { s_rf[o*256+tid] = tb_iou[o]; s_ri[o*256+tid] = tb_pri[o]; }
  __syncthreads();
  if (tid < NO) {
    float bi = -1.f; int bp = 0x7fffffff;
    for (int t = 0; t < 256; ++t) {
      const float v = s_rf[tid*256+t]; const int pr = s_ri[tid*256+t];
      if (v > bi || (v == bi && pr < bp)) { bi = v; bp = pr; }
    }
    s_objprior[tid] = bp;
  }
  __syncthreads();

  // ---- pass 2: per-prior argmax over objects + forced-match override ----
  int cnt = 0; float l1 = 0.f;
  for (int p = tid; p < NP; p += 256) {
    const float4 pr = pri4[p];
    const float px0 = pr.x - 0.5f*pr.z, py0 = pr.y - 0.5f*pr.w;
    const float px1 = pr.x + 0.5f*pr.z, py1 = pr.y + 0.5f*pr.w;
    const float pa = pr.z * pr.w;
    float biou = -1.f; int bo = 0;
#pragma unroll
    for (int o = 0; o < NO; ++o) {
      const float lx = fmaxf(s_box[o][0], px0), ly = fmaxf(s_box[o][1], py0);
      const float rx = fminf(s_box[o][2], px1), ry = fminf(s_box[o][3], py1);
      const float iw = fmaxf(rx - lx, 0.f), ih = fmaxf(ry - ly, 0.f);
      const float inter = iw * ih;
      const float iou = inter / (s_area[o] + pa - inter);
      if (iou > biou) { biou = iou; bo = o; }
    }
#pragma unroll
    for (int o = 0; o < NO; ++o)       // later object wins on duplicates
      if (s_objprior[o] == p) { bo = o; biou = 1.0f; }

    const int lab = (biou < 0.5f) ? 0 : s_lab[bo];
    labels_ws[b * NP + p] = lab;
    if (lab != 0) {
      ++cnt;
      const float bx0 = s_box[bo][0], by0 = s_box[bo][1];
      const float bx1 = s_box[bo][2], by1 = s_box[bo][3];
      const float g0 = ((bx0 + bx1) * 0.5f - pr.x) / (pr.z * VAR_XY);
      const float g1 = ((by0 + by1) * 0.5f - pr.y) / (pr.w * VAR_XY);
      const float g2 = logf((bx1 - bx0) / pr.z) / VAR_WH;
      const float g3 = logf((by1 - by0) / pr.w) / VAR_WH;
      const float4 pl = pl4[(size_t)b * NP + p];
      l1 += fabsf(pl.x-g0) + fabsf(pl.y-g1) + fabsf(pl.z-g2) + fabsf(pl.w-g3);
    }
  }
  __syncthreads();
  s_rf[tid] = l1; s_ri[tid] = cnt;
  __syncthreads();
  if (tid == 0) {
    float L = 0.f; int C = 0;
    for (int t = 0; t < 256; ++t) { L += s_rf[t]; C += s_ri[t]; }
    l1_part[b] = L; n_pos[b] = C;
  }
}

// ---------------------------------------------------------------------------
// Kernel 2: cross-entropy per prior.  One wave32 per 16-prior tile.
// Tile staged to LDS via TDM (tensor_load_to_lds) when available, else b128
// loads; max via shuffle; sum-of-exp via V_WMMA_F32_16X16X4_F32 (B = ones).
// ---------------------------------------------------------------------------
__global__ __launch_bounds__(256) void mbx_ce(
    const float* __restrict__ confs,     // [B,P,C]
    const int*   __restrict__ labels_ws, // [B,P]
    float*       __restrict__ ce_all)    // [B,P]
{
  const int b    = blockIdx.y;
  const int wave = threadIdx.x >> 5;
  const int lane = threadIdx.x & 31;
  const int p0   = (blockIdx.x * 8 + wave) * 16;
  __shared__ __align__(16) float s_log[8][16 * NC];   // 41.5 KB, 16B aligned slices
  __shared__ float s_sum[8][16];
  if (p0 >= NP) return;                  // wave-uniform

  float* sl = &s_log[wave][0];
  const long long base   = ((long long)b * NP + p0) * NC;   // multiple of 4
  const long long totalf = (long long)NB * NP * NC;

#if __has_builtin(__builtin_amdgcn_tensor_load_to_lds) && __has_builtin(__builtin_amdgcn_s_wait_tensorcnt)
  // --- Tensor Data Mover: one 5184 B contiguous tile -> LDS per wave. ---
  // D# packing per CDNA5 ISA 8.3/8.4. tensor_dim0 clamped to buffer end so
  // OOB reads of the tail tile return zero. EXEC ignored; one op per wave.
  {
    const unsigned long long gaddr =
        (unsigned long long)(const void*)confs + (unsigned long long)base * 4ull;
    const long long remain = totalf - base;
    const unsigned td0 = (unsigned)((remain < (long long)(16 * NC)) ? remain : (16 * NC));
    const unsigned g_lo  = (unsigned)__builtin_amdgcn_readfirstlane((int)(unsigned)(gaddr & 0xffffffffull));
    const unsigned g_hi  = (unsigned)__builtin_amdgcn_readfirstlane((int)(unsigned)(gaddr >> 32));
    const unsigned ldsa  = (unsigned)__builtin_amdgcn_readfirstlane((int)(unsigned)(unsigned long long)(void*)sl);
    const unsigned tdim0 = (unsigned)__builtin_amdgcn_readfirstlane((int)td0);

    u32x4 g0;
    g0[0] = 1u;                                   // count=1, user descriptor
    g0[1] = ldsa;                                 // lds_addr (bytes)
    g0[2] = g_lo;                                 // global_addr[31:0]
    g0[3] = (g_hi & 0x01ffffffu) | (2u << 30);    // global_addr[56:32] | type=2

    i32x8 g1;
    g1[0] = (int)(2u << 16);                                  // data_size=4B
    g1[1] = (int)((tdim0 & 0xffffu) << 16);                   // tensor_dim0[15:0]
    g1[2] = (int)(((tdim0 >> 16) & 0xffffu) | (1u << 16));    // tensor_dim0[31:16] | tensor_dim1=1
    g1[3] = (int)((unsigned)(16 * NC) << 16);                 // tile_dim0 = 1296
    g1[4] = (int)1u;                                          // tile_dim1=1, tile_dim2=0
    g1[5] = (int)(16 * NC);                                   // tensor_dim0_stride lo
    g1[6] = (int)((unsigned)(16 * NC) << 16);                 // dim0_stride hi=0 | dim1_stride[15:0]
    g1[7] = 0;                                                // dim1_stride hi

    const i32x4 gz4 = {0, 0, 0, 0};               // <=2D: groups 2/3 unused
    const i32x8 gz8 = {0, 0, 0, 0, 0, 0, 0, 0};   // 6-arg form: extra group
    __builtin_amdgcn_tensor_load_to_lds(g0, g1, gz4, gz4, gz8, 0);
    __builtin_amdgcn_s_wait_tensorcnt(0);
    asm volatile("" ::: "memory");
  }
#else
  // --- fallback: coalesced b128 staging (324 float4 per tile) ---
  {
    const float4* g4 = (const float4*)confs;
    const long long base4 = base >> 2;
    const long long max4  = (totalf >> 2) - 1;
    for (int i = lane; i < (16 * NC) / 4; i += 32) {
      long long q = base4 + i; if (q > max4) q = max4;   // clamp tail tile
      ((float4*)sl)[i] = g4[q];
    }
    asm volatile("s_wait_dscnt 0" ::: "memory");
  }
#endif

  // row max: lane handles row r = lane%16, classes c = h, h+2, ...
  const int r = lane & 15, h = lane >> 4;
  float m = -3.4e38f;
  for (int c = h; c < NC; c += 2) m = fmaxf(m, sl[r * NC + c]);
  m = fmaxf(m, __shfl_xor(m, 16));

  // WMMA exp-sum: A (16x4 f32) lane layout: row = lane%16, K-pair = lane/16.
  v8f acc = {0.f,0.f,0.f,0.f,0.f,0.f,0.f,0.f};
  const v2f ones = {1.0f, 1.0f};
#pragma unroll
  for (int k = 0; k < 21; ++k) {
    const int c0 = 4 * k + 2 * h;
    v2f a;
    a.x = (c0     < NC) ? expf(sl[r * NC + c0    ] - m) : 0.f;
    a.y = (c0 + 1 < NC) ? expf(sl[r * NC + c0 + 1] - m) : 0.f;
    acc = __builtin_amdgcn_wmma_f32_16x16x4_f32(
        /*neg_a=*/false, a, /*neg_b=*/false, ones,
        /*c_mod=*/(short)0, acc, /*reuse_a=*/false, /*reuse_b=*/false);
  }

  // D layout: lane holds rows (v + 8*(lane/16)) at col lane%16 (all cols equal).
  if ((lane & 15) == 0) {
    const int half = lane >> 4;
#pragma unroll
    for (int v = 0; v < 8; ++v) s_sum[wave][half * 8 + v] = acc[v];
  }
  asm volatile("s_wait_dscnt 0" ::: "memory");

  if (lane < 16) {                        // lane == row, h == 0, m == row max
    const int p = p0 + lane;
    if (p < NP) {
      const int   lab = labels_ws[b * NP + p];
      const float x   = sl[lane * NC + lab];
      const float ce  = m + logf(s_sum[wave][lane]) - x;
      ce_all[b * NP + p] = fmaxf(ce, 0.f);
    }
  }
}

// ---------------------------------------------------------------------------
// Kernel 3: per-batch hard-negative mining: exact sum of top-K CE values
// (K = 3*n_pos) via bit-wise threshold search over the LDS-resident array.
// ---------------------------------------------------------------------------
__global__ __launch_bounds__(256) void mbx_hard(
    const float* __restrict__ ce_all,
    const int*   __restrict__ labels_ws,
    const int*   __restrict__ n_pos,
    float*       __restrict__ conf_part)  // [B] = ce_pos_sum + ce_hard_sum
{
  const int b = blockIdx.x, tid = threadIdx.x;
  __shared__ float s_ce[NP];              // 34.9 KB of 320 KB WGP LDS
  __shared__ float s_rf[256];
  __shared__ int   s_ri[256];
  __shared__ float s_bf[2];
  __shared__ int   s_bi[2];

  float psum = 0.f, tsum = 0.f; int c0 = 0;
  for (int p = tid; p < NP; p += 256) {
    const float ce  = ce_all[b * NP + p];
    const int   lab = labels_ws[b * NP + p];
    const float cn  = (lab != 0) ? 0.f : ce;
    if (lab != 0) psum += ce;
    s_ce[p] = cn;
    tsum += cn;
    if (cn > 0.f) ++c0;
  }
  s_rf[tid] = psum; s_ri[tid] = c0;
  __syncthreads();
  if (tid == 0) {
    float a = 0.f; int c = 0;
    for (int t = 0; t < 256; ++t) { a += s_rf[t]; c += s_ri[t]; }
    s_bf[0] = a; s_bi[0] = c;
  }
  __syncthreads();
  const float pos_total = s_bf[0];
  const int   C0        = s_bi[0];
  s_rf[tid] = tsum;
  __syncthreads();
  if (tid == 0) { float a = 0.f; for (int t = 0; t < 256; ++t) a += s_rf[t]; s_bf[1] = a; }
  __syncthreads();
  const float neg_total = s_bf[1];

  const int K = 3 * n_pos[b];
  if (K <= 0) {
    if (tid == 0) conf_part[b] = pos_total;
    return;
  }
  if (C0 <= K) {  // top-K contains every nonzero value (zeros contribute 0)
    if (tid == 0) conf_part[b] = pos_total + neg_total;
    return;
  }

  // Values are nonneg floats -> uint bit pattern order == value order.
  // Find max t with count(v > float(t)) >= K; then K-th largest = t+1.
  unsigned t = 0u;
  for (int bit = 30; bit >= 0; --bit) {
    const unsigned cand = t | (1u << bit);
    const float tv = __uint_as_float(cand);
    int c = 0;
    for (int p = tid; p < NP; p += 256) c += (s_ce[p] > tv) ? 1 : 0;
    s_ri[tid] = c;
    __syncthreads();
    if (tid == 0) { int a = 0; for (int q = 0; q < 256; ++q) a += s_ri[q]; s_bi[1] = a; }
    __syncthreads();
    if (s_bi[1] >= K) t = cand;
  }
  const float fstar = __uint_as_float(t + 1u);   // K-th largest value
  float sgt = 0.f; int cgt = 0;
  for (int p = tid; p < NP; p += 256) {
    const float v = s_ce[p];
    if (v > fstar) { sgt += v; ++cgt; }
  }
  s_rf[tid] = sgt; s_ri[tid] = cgt;
  __syncthreads();
  if (tid == 0) {
    float a = 0.f; int c = 0;
    for (int q = 0; q < 256; ++q) { a += s_rf[q]; c += s_ri[q]; }
    conf_part[b] = pos_total + a + (float)(K - c) * fstar;
  }
}

// ---------------------------------------------------------------------------
// Kernel 4: combine (fixed-order, deterministic).
// ---------------------------------------------------------------------------
__global__ void mbx_final(const int* __restrict__ n_pos,
                          const float* __restrict__ l1_part,
                          const float* __restrict__ conf_part,
                          float* __restrict__ out)
{
  float npt = 0.f, l1 = 0.f, cf = 0.f;
  for (int b = 0; b < NB; ++b) {
    npt += (float)n_pos[b]; l1 += l1_part[b]; cf += conf_part[b];
  }
  out[0] = cf / npt + l1 / (npt * 4.0f);
}

extern "C" void kernel_launch(void* const* d_in, const int* in_sizes, int n_in,
                              void* d_out, int out_size, void* d_ws, size_t ws_size,
                              hipStream_t stream)
{
  const float* plocs   = (const float*)d_in[0];  // [B,P,4]
  const float* pconfs  = (const float*)d_in[1];  // [B,P,C]
  const float* tboxes  = (const float*)d_in[2];  // [B,O,4]
  const int*   tlabels = (const int*)  d_in[3];  // [B,O]
  const float* priors  = (const float*)d_in[4];  // [P,4]
  float* out = (float*)d_out;

  char* ws = (char*)d_ws;
  int*   labels_ws = (int*)ws;    ws += (size_t)NB * NP * sizeof(int);
  float* ce_all    = (float*)ws;  ws += (size_t)NB * NP * sizeof(float);
  int*   npos      = (int*)ws;    ws += (size_t)NB * sizeof(int);
  float* l1_part   = (float*)ws;  ws += (size_t)NB * sizeof(float);
  float* conf_part = (float*)ws;  ws += (size_t)NB * sizeof(float);

  mbx_match<<<NB, 256, 0, stream>>>(plocs, tboxes, tlabels, priors,
                                    labels_ws, npos, l1_part);
  dim3 gce((NP + 127) / 128, NB);  // 8 waves/block, 16 priors per wave
  mbx_ce<<<gce, 256, 0, stream>>>(pconfs, labels_ws, ce_all);
  mbx_hard<<<NB, 256, 0, stream>>>(ce_all, labels_ws, npos, conf_part);
  mbx_final<<<1, 1, 0, stream>>>(npos, l1_part, conf_part, out);
}